// GNNMovieLensModel_2216203125485
// MI455X (gfx1250) — compile-verified
//
#include <hip/hip_runtime.h>
#include <hip/hip_bf16.h>

#define NUM_MOVIES 60000
#define NUM_USERS  200000
#define N_NODES    260000       // multiple of 16 -> 16250 row tiles
#define EMB        64
#define HID        128
#define GEN        20
#define NEDGE      2000000
#define NBATCH     100000       // multiple of 16 -> 6250 row tiles
#define KCOMB      (2 * HID + GEN)   // 276 = 3 * 92, multiple of 4
// LDS pitch per K-row-pair (floats): 2*HID + 32; 288 mod 64 == 32 -> the two
// half-wave row-pairs hit disjoint 32-bank halves: conflict-free ds_load_b64.
#define PPITCH     288

typedef float v2f __attribute__((ext_vector_type(2)));
typedef float v8f __attribute__((ext_vector_type(8)));

// ----------------------------------------------------------------------------
// degree / normalization
// ----------------------------------------------------------------------------
__global__ void k_deg_init(float* deg) {
    int i = blockIdx.x * 256 + threadIdx.x;
    if (i < N_NODES) deg[i] = 1.0f;            // self-loop contributes 1
}

__global__ void k_deg_accum(const int* __restrict__ dst, float* deg) {
    int e = blockIdx.x * 256 + threadIdx.x;
    if (e < NEDGE) atomicAdd(&deg[dst[e]], 1.0f);
}

__global__ void k_dinv(float* deg) {
    int i = blockIdx.x * 256 + threadIdx.x;
    if (i < N_NODES) deg[i] = rsqrtf(deg[i]);  // deg >= 1 always
}

// ----------------------------------------------------------------------------
// Cooperative W-chunk load into pair-interleaved LDS:
//   Wl[(r>>1)*PPITCH + col*2 + (r&1)] = W[k0+r][col]
// Each thread moves a 2x2 block: two global_load_b64 -> one ds_store_b128.
// ----------------------------------------------------------------------------
__device__ __forceinline__ void load_w_chunk(float* Wl, const float* __restrict__ W,
                                             int k0, int KC) {
    const int nblk = (KC / 2) * (HID / 2);
    for (int idx = threadIdx.x; idx < nblk; idx += 256) {
        int rp = idx >> 6;                 // / (HID/2)
        int c2 = (idx & 63) << 1;
        const float* wp = W + (size_t)(k0 + rp * 2) * HID + c2;
        float2 w0 = *(const float2*)(wp);
        float2 w1 = *(const float2*)(wp + HID);
        float4 v;  v.x = w0.x; v.y = w1.x; v.z = w0.y; v.w = w1.y;
        *(float4*)(&Wl[rp * PPITCH + (c2 << 1)]) = v;
    }
}

// ----------------------------------------------------------------------------
// Register-blocked WMMA fp32 GEMM: OUT[M x 128] = act(A) @ W.
// Block = 8 waves; each wave owns ONE 16-row tile across ALL 128 columns
// (8 independent 16x16 accumulator chains) -> A fragments loaded once.
// B fragments: single aligned ds_load_b64 from pair-interleaved LDS.
// ----------------------------------------------------------------------------
template <int K, int KC, bool RELU, bool CONCAT>
__global__ __launch_bounds__(256) void k_gemm_rb(const float* __restrict__ A,
                                                 const float* __restrict__ A2,
                                                 const float* __restrict__ W,
                                                 float* __restrict__ OUT,
                                                 int ntiles) {
    __shared__ float Wl[(KC / 2) * PPITCH];

    const int wave  = threadIdx.x >> 5;
    const int lane  = threadIdx.x & 31;
    const int half  = lane >> 4;          // 0: K pair {0,1}, 1: {2,3}
    const int l16   = lane & 15;
    const int koff  = half * 2;
    const int mtile = blockIdx.x * 8 + wave;
    const bool valid = (mtile < ntiles);

    const float* arow = nullptr;
    if (valid) {
        int row = mtile * 16 + l16;
        if (CONCAT)
            arow = (row < NUM_MOVIES) ? (A + (size_t)row * K)
                                      : (A2 + (size_t)(row - NUM_MOVIES) * K);
        else
            arow = A + (size_t)row * K;
    }

    v8f acc[8] = {};

    for (int k0 = 0; k0 < K; k0 += KC) {
        load_w_chunk(Wl, W, k0, KC);
        __syncthreads();

        if (valid) {
            const float* ar = arow + k0;
            #pragma unroll
            for (int kk = 0; kk < KC; kk += 4) {
                float a0 = ar[kk + koff];
                float a1 = ar[kk + koff + 1];
                if (RELU) { a0 = fmaxf(a0, 0.0f); a1 = fmaxf(a1, 0.0f); }
                v2f a; a[0] = a0; a[1] = a1;
                const int rp = (kk + koff) >> 1;   // row-pair index in LDS
                #pragma unroll
                for (int t = 0; t < 8; ++t) {
                    v2f b = *(const v2f*)(&Wl[rp * PPITCH + ((t * 16 + l16) << 1)]);
                    acc[t] = __builtin_amdgcn_wmma_f32_16x16x4_f32(
                        false, a, false, b, (short)0, acc[t], false, false);
                }
            }
        }
        __syncthreads();
    }

    if (valid) {
        const int orow = mtile * 16 + half * 8;
        #pragma unroll
        for (int t = 0; t < 8; ++t) {
            float* o = OUT + (size_t)orow * HID + t * 16 + l16;
            #pragma unroll
            for (int r = 0; r < 8; ++r) o[(size_t)r * HID] = acc[t][r];
        }
    }
}

// ----------------------------------------------------------------------------
// MLP head GEMM: relu(COMB @ fc1_w + fc1_b) with fc2 (128 -> 1) fully fused.
// Lane-local fc2 partial dot -> __shfl_xor butterfly within each 16-lane half
// -> plain store of out[row] (no atomics, no init kernel).
// ----------------------------------------------------------------------------
__global__ __launch_bounds__(256) void k_gemm_mlp(const float* __restrict__ COMB,
                                                  const float* __restrict__ W,
                                                  const float* __restrict__ b1v,
                                                  const float* __restrict__ fc2w,
                                                  const float* __restrict__ fc2b,
                                                  float* __restrict__ out,
                                                  int ntiles) {
    const int KC = 92;                    // 276 = 3 * 92
    __shared__ float Wl[(KC / 2) * PPITCH];

    const int wave  = threadIdx.x >> 5;
    const int lane  = threadIdx.x & 31;
    const int half  = lane >> 4;
    const int l16   = lane & 15;
    const int koff  = half * 2;
    const int mtile = blockIdx.x * 8 + wave;
    const bool valid = (mtile < ntiles);
    const float* arow = valid ? (COMB + (size_t)(mtile * 16 + l16) * KCOMB) : COMB;

    v8f acc[8] = {};

    for (int k0 = 0; k0 < KCOMB; k0 += KC) {
        load_w_chunk(Wl, W, k0, KC);
        __syncthreads();

        if (valid) {
            const float* ar = arow + k0;
            #pragma unroll
            for (int kk = 0; kk < KC; kk += 4) {
                v2f a; a[0] = ar[kk + koff]; a[1] = ar[kk + koff + 1];
                const int rp = (kk + koff) >> 1;
                #pragma unroll
                for (int t = 0; t < 8; ++t) {
                    v2f b = *(const v2f*)(&Wl[rp * PPITCH + ((t * 16 + l16) << 1)]);
                    acc[t] = __builtin_amdgcn_wmma_f32_16x16x4_f32(
                        false, a, false, b, (short)0, acc[t], false, false);
                }
            }
        }
        __syncthreads();
    }

    if (valid) {
        float rowsum[8];
        #pragma unroll
        for (int r = 0; r < 8; ++r) rowsum[r] = 0.0f;
        #pragma unroll
        for (int t = 0; t < 8; ++t) {
            float bias = b1v[t * 16 + l16];
            float w2   = fc2w[t * 16 + l16];       // fc2_w shape [HID, 1]
            #pragma unroll
            for (int r = 0; r < 8; ++r)
                rowsum[r] += fmaxf(acc[t][r] + bias, 0.0f) * w2;
        }
        #pragma unroll
        for (int m = 8; m >= 1; m >>= 1) {
            #pragma unroll
            for (int r = 0; r < 8; ++r)
                rowsum[r] += __shfl_xor(rowsum[r], m);
        }
        if (l16 == 0) {
            const float b0 = fc2b[0];
            const int orow = mtile * 16 + half * 8;
            #pragma unroll
            for (int r = 0; r < 8; ++r) out[orow + r] = rowsum[r] + b0;
        }
    }
}

// ----------------------------------------------------------------------------
// AGG[n][c] = H[n][c] * dinv[n]^2 + bias[c]   (self-loop message + bias)
// ----------------------------------------------------------------------------
__global__ void k_agg_init(const float* __restrict__ H,
                           const float* __restrict__ dinv,
                           const float* __restrict__ bias,
                           float* __restrict__ AGG) {
    int idx = blockIdx.x * 256 + threadIdx.x;
    if (idx < N_NODES * HID) {
        int n = idx >> 7;
        int c = idx & (HID - 1);
        float di = dinv[n];
        AGG[idx] = H[idx] * di * di + bias[c];
    }
}

// ----------------------------------------------------------------------------
// Edge scatter: AGG[dst] += H[src] * (dinv[src]*dinv[dst]).  One wave/edge,
// 4 channels/lane; gathers + atomics are L2-resident (133 MB < 192 MB L2).
// ----------------------------------------------------------------------------
__global__ void k_scatter(const int* __restrict__ src, const int* __restrict__ dst,
                          const float* __restrict__ dinv,
                          const float* __restrict__ H, float* __restrict__ AGG) {
    int e = blockIdx.x * 8 + (threadIdx.x >> 5);
    if (e >= NEDGE) return;
    int lane = threadIdx.x & 31;
    int s = src[e], d = dst[e];
    float w = dinv[s] * dinv[d];
    int c = lane << 2;
    const float4 hv = *(const float4*)(H + (size_t)s * HID + c);
    float* ap = AGG + (size_t)d * HID + c;
    atomicAdd(ap + 0, hv.x * w);
    atomicAdd(ap + 1, hv.y * w);
    atomicAdd(ap + 2, hv.z * w);
    atomicAdd(ap + 3, hv.w * w);
}

// ----------------------------------------------------------------------------
// COMB[i] = [ X[movie_id[i]] | X[user_id[i]+NUM_MOVIES] | genre_f32 ]
// ----------------------------------------------------------------------------
__global__ void k_gather(const int* __restrict__ mid, const int* __restrict__ uid,
                         const int* __restrict__ gen,
                         const float* __restrict__ X, float* __restrict__ COMB) {
    int i = blockIdx.x;
    int t = threadIdx.x;                   // 0..127
    float* crow = COMB + (size_t)i * KCOMB;
    int m = mid[i];
    int u = uid[i] + NUM_MOVIES;
    crow[t]       = X[(size_t)m * HID + t];
    crow[HID + t] = X[(size_t)u * HID + t];
    if (t < GEN) crow[2 * HID + t] = (float)gen[i * GEN + t];
}

// ----------------------------------------------------------------------------
extern "C" void kernel_launch(void* const* d_in, const int* in_sizes, int n_in,
                              void* d_out, int out_size, void* d_ws, size_t ws_size,
                              hipStream_t stream) {
    const int*   movie_id = (const int*)  d_in[0];
    const int*   user_id  = (const int*)  d_in[1];
    const int*   genre_id = (const int*)  d_in[2];
    const int*   eidx     = (const int*)  d_in[3];
    const float* memb     = (const float*)d_in[4];
    const float* uemb     = (const float*)d_in[5];
    const float* W1       = (const float*)d_in[6];
    const float* b1       = (const float*)d_in[7];
    const float* W2       = (const float*)d_in[8];
    const float* b2       = (const float*)d_in[9];
    const float* fc1w     = (const float*)d_in[10];
    const float* fc1b     = (const float*)d_in[11];
    const float* fc2w     = (const float*)d_in[12];
    const float* fc2b     = (const float*)d_in[13];
    float* out = (float*)d_out;

    const int* src = eidx;            // edge_index row 0
    const int* dst = eidx + NEDGE;    // edge_index row 1

    // workspace layout (floats): dinv[N] | bufA[N*HID] | bufB[N*HID]
    float* dinv = (float*)d_ws;
    float* bufA = dinv + N_NODES;                     // H1 -> H2 -> COMB
    float* bufB = bufA + (size_t)N_NODES * HID;       // AGG1 -> AGG2 (final x)

    const int nodeBlocks = (N_NODES + 255) / 256;
    const int edgeBlocks = (NEDGE + 255) / 256;
    const int elemBlocks = (N_NODES * HID + 255) / 256;
    const int NT_NODE  = N_NODES / 16;                // 16250
    const int NT_BATCH = NBATCH / 16;                 // 6250

    // ---- degree / normalization ----
    k_deg_init <<<nodeBlocks, 256, 0, stream>>>(dinv);
    k_deg_accum<<<edgeBlocks, 256, 0, stream>>>(dst, dinv);
    k_dinv     <<<nodeBlocks, 256, 0, stream>>>(dinv);

    // ---- GCN layer 1: H1 = concat(emb) @ W1 ----
    k_gemm_rb<EMB, EMB, false, true>
        <<<(NT_NODE + 7) / 8, 256, 0, stream>>>(memb, uemb, W1, bufA, NT_NODE);
    k_agg_init<<<elemBlocks, 256, 0, stream>>>(bufA, dinv, b1, bufB);     // AGG1
    k_scatter <<<NEDGE / 8, 256, 0, stream>>>(src, dst, dinv, bufA, bufB);

    // ---- GCN layer 2: H2 = relu(AGG1) @ W2 (relu fused in A-load) ----
    k_gemm_rb<HID, 64, true, false>
        <<<(NT_NODE + 7) / 8, 256, 0, stream>>>(bufB, nullptr, W2, bufA, NT_NODE);
    k_agg_init<<<elemBlocks, 256, 0, stream>>>(bufA, dinv, b2, bufB);     // AGG2
    k_scatter <<<NEDGE / 8, 256, 0, stream>>>(src, dst, dinv, bufA, bufB);

    // ---- MLP head (fc1 + relu + fc2 fully fused, no atomics) ----
    k_gather  <<<NBATCH, HID, 0, stream>>>(movie_id, user_id, genre_id,
                                           bufB, bufA);                   // COMB
    k_gemm_mlp<<<(NT_BATCH + 7) / 8, 256, 0, stream>>>(bufA, fc1w, fc1b,
                                                       fc2w, fc2b, out, NT_BATCH);
}